// SimGCL_1683627180409
// MI455X (gfx1250) — compile-verified
//
#include <hip/hip_runtime.h>

#define EMBED 64

typedef float v2f __attribute__((ext_vector_type(2)));
typedef float v8f __attribute__((ext_vector_type(8)));

// acc = 0.25 * concat(playlist, track); zbuf = 0 (first scatter destination).
__global__ void k_init(const float4* __restrict__ pl, const float4* __restrict__ tr,
                       float4* __restrict__ acc, float4* __restrict__ zbuf,
                       int n4, int p4) {
    int i = blockIdx.x * blockDim.x + threadIdx.x;
    if (i >= n4) return;
    float4 v = (i < p4) ? pl[i] : tr[i - p4];
    float4 s;
    s.x = 0.25f * v.x; s.y = 0.25f * v.y; s.z = 0.25f * v.z; s.w = 0.25f * v.w;
    acc[i] = s;
    zbuf[i] = make_float4(0.f, 0.f, 0.f, 0.f);
}

// One edge per 16-lane group: gather 64 floats (float4/lane), scale by edge
// weight, scatter with hardware f32 atomics (no-return -> global_atomic_add_f32).
// Edge arrays are streamed once -> non-temporal loads keep them out of the L2
// working set (the 153.6 MB embedding table fits in the 192 MB L2).
__global__ void k_scatter(const float* __restrict__ cur, float* __restrict__ nxt,
                          const float* __restrict__ ew, const int* __restrict__ es,
                          const int* __restrict__ ed, int nnz) {
    int t = blockIdx.x * blockDim.x + threadIdx.x;
    int e = t >> 4;
    int lane = t & 15;
    if (e >= nnz) return;
    float w = __builtin_nontemporal_load(ew + e);
    int   s = __builtin_nontemporal_load(es + e);
    int   d = __builtin_nontemporal_load(ed + e);
    const float4* srow = (const float4*)(cur + (size_t)s * EMBED);
    float4 x = srow[lane];
    float* drow = nxt + (size_t)d * EMBED + (lane << 2);
    __hip_atomic_fetch_add(drow + 0, w * x.x, __ATOMIC_RELAXED, __HIP_MEMORY_SCOPE_AGENT);
    __hip_atomic_fetch_add(drow + 1, w * x.y, __ATOMIC_RELAXED, __HIP_MEMORY_SCOPE_AGENT);
    __hip_atomic_fetch_add(drow + 2, w * x.z, __ATOMIC_RELAXED, __HIP_MEMORY_SCOPE_AGENT);
    __hip_atomic_fetch_add(drow + 3, w * x.w, __ATOMIC_RELAXED, __HIP_MEMORY_SCOPE_AGENT);
}

// Per 16x16 tile (one wave32 each): acc_tile = Y_tile x (0.25*I) + acc_tile via
// four V_WMMA_F32_16X16X4_F32 K-slices (C-matrix = accumulator, exact math since
// 0.25*I contributes one product per output). Also zeroes the retired buffer Z
// so it can serve as the next layer's scatter destination.
__global__ void k_post_wmma(const float* __restrict__ Y, float* __restrict__ acc,
                            float* __restrict__ Z, int ntiles) {
    int wave = blockIdx.x * (blockDim.x >> 5) + (threadIdx.x >> 5);
    if (wave >= ntiles) return;                 // wave-uniform: EXEC stays all-ones
    int lane = threadIdx.x & 31;
    int half = lane >> 4;                       // 0: lanes 0-15, 1: lanes 16-31
    int l16  = lane & 15;
    int r0 = (wave >> 2) << 4;                  // row tile (EMBED/16 == 4 col tiles)
    int c0 = (wave & 3) << 4;

    // C/D 16x16 f32 layout: VGPR v, lane L -> (M = v + 8*(L>>4), N = L&15)
    size_t cbase = (size_t)(r0 + (half << 3)) * EMBED + c0 + l16;
    v8f c;
#pragma unroll
    for (int v = 0; v < 8; ++v) c[v] = acc[cbase + (size_t)v * EMBED];

    // A 16x4 f32 layout: lane L -> M = L&15; K = 2*(L>>4) + {0,1} in VGPR {0,1}
    const float* yrow = Y + (size_t)(r0 + l16) * EMBED + c0 + (half << 1);
#pragma unroll
    for (int kb = 0; kb < 4; ++kb) {
        v2f a;
        a.x = yrow[4 * kb];
        a.y = yrow[4 * kb + 1];
        // B 4x16 = rows [4kb..4kb+4) of 0.25*I_16 (mirrored A layout: lane = N,
        // K = 2*(L>>4) + {0,1}).
        int kk = 4 * kb + (half << 1);
        v2f b;
        b.x = (l16 == kk)     ? 0.25f : 0.0f;
        b.y = (l16 == kk + 1) ? 0.25f : 0.0f;
        c = __builtin_amdgcn_wmma_f32_16x16x4_f32(false, a, false, b,
                                                  (short)0, c, false, false);
    }

#pragma unroll
    for (int v = 0; v < 8; ++v) {
        acc[cbase + (size_t)v * EMBED] = c[v];
        Z[cbase + (size_t)v * EMBED] = 0.0f;
    }
}

extern "C" void kernel_launch(void* const* d_in, const int* in_sizes, int n_in,
                              void* d_out, int out_size, void* d_ws, size_t ws_size,
                              hipStream_t stream) {
    const float* pl = (const float*)d_in[0];
    const float* tr = (const float*)d_in[1];
    const float* ew = (const float*)d_in[2];
    const int*   es = (const int*)d_in[3];
    const int*   ed = (const int*)d_in[4];

    int pElems = in_sizes[0];            // NUM_PLAYLISTS * 64
    int tElems = in_sizes[1];            // NUM_TRACKS * 64
    int nnz    = in_sizes[2];
    int nElems = pElems + tElems;        // NUM_NODES * 64
    int nNodes = nElems / EMBED;

    float* acc  = (float*)d_out;                     // running 0.25-scaled sum
    float* bufA = (float*)d_ws;                      // ping-pong layer buffers
    float* bufB = bufA + (size_t)nElems;

    // init: acc = 0.25*emb, bufB = 0 (first scatter target)
    int n4 = nElems / 4, p4 = pElems / 4;
    k_init<<<(n4 + 255) / 256, 256, 0, stream>>>(
        (const float4*)pl, (const float4*)tr, (float4*)acc, (float4*)bufB, n4, p4);

    // cur = concat(playlist, track)
    hipMemcpyAsync(bufA, pl, (size_t)pElems * sizeof(float),
                   hipMemcpyDeviceToDevice, stream);
    hipMemcpyAsync(bufA + pElems, tr, (size_t)tElems * sizeof(float),
                   hipMemcpyDeviceToDevice, stream);

    int ntiles = (nNodes / 16) * (EMBED / 16);       // 37500 * 4 = 150000
    int postBlocks = (ntiles + 7) / 8;               // 8 waves / 256-thread block
    long long sthreads = (long long)nnz * 16;
    int sblocks = (int)((sthreads + 255) / 256);

    float* cur = bufA;
    float* nxt = bufB;
    for (int l = 0; l < 3; ++l) {
        k_scatter<<<sblocks, 256, 0, stream>>>(cur, nxt, ew, es, ed, nnz);
        k_post_wmma<<<postBlocks, 256, 0, stream>>>(nxt, acc, cur, ntiles);
        float* tmp = cur; cur = nxt; nxt = tmp;      // old cur (now zeroed) is next target
    }
}